// ROIPooler_74156905333241
// MI455X (gfx1250) — compile-verified
//
#include <hip/hip_runtime.h>
#include <math.h>

// FPN ROIAlign (OUT=7, SR=2, C=256, levels 2..5, scales 0.25/0.125/0.0625/0.03125)
// Stage A: gather + x-interpolate 28 tap rows -> LDS P[c][28][7]
// Stage B: y-reduction + bin average via chained V_WMMA_F32_16X16X4_F32

#define POUT 7
#define PSR  2
#define NS   14            // sample positions per axis (OUT*SR)
#define NTAP 28            // tap rows (2 per y-sample, with duplicates)
#define CT   64            // channels per block
#define PSTR 197           // 28*7 + 1 padding (conflict-free LDS stride)

typedef float v2f __attribute__((ext_vector_type(2)));
typedef float v8f __attribute__((ext_vector_type(8)));

__global__ __launch_bounds__(256) void roi_align_fpn_wmma(
    const float* __restrict__ f0, const float* __restrict__ f1,
    const float* __restrict__ f2, const float* __restrict__ f3,
    const float* __restrict__ rois, float* __restrict__ out, int M)
{
    __shared__ int   sXL[NS], sXH[NS];
    __shared__ float sWHX[NS], sWLX[NS];
    __shared__ int   sYROW[NTAP];
    __shared__ float sWY[NTAP];
    __shared__ int   sMeta[3];            // level idx (0..3), batch, W(==H)
    __shared__ float sP[CT * PSTR];       // ~50.4 KB

    const int m  = blockIdx.x;
    const int ct = blockIdx.y;            // channel tile: 0..(256/CT - 1)
    const int t  = threadIdx.x;

    // ---- per-ROI geometry (recomputed by the first 29 threads; cheap) ----
    if (t < 29) {
        const float* roi = rois + (size_t)m * 5;
        float rx1 = roi[1], ry1 = roi[2], rx2 = roi[3], ry2 = roi[4];
        float rw = rx2 - rx1, rh = ry2 - ry1;
        float size = sqrtf(rw * rh);
        int lvl = (int)floorf(4.0f + log2f(size * (1.0f / 224.0f) + 1.1920929e-07f));
        lvl = lvl < 2 ? 2 : (lvl > 5 ? 5 : lvl);
        int l = lvl - 2;
        float scale = 0.25f / (float)(1 << l);
        int Wd = 256 >> l;                 // feature H == W
        if (t < NS) {
            // x sample t
            int s = t;
            float x1 = rx1 * scale - 0.5f, x2 = rx2 * scale - 0.5f;
            float g  = ((float)s + 0.5f) * (1.0f / PSR);
            float x  = x1 + g * (x2 - x1) * (1.0f / POUT);
            float mf = ((x > -1.0f) && (x < (float)Wd)) ? 1.0f : 0.0f;
            float xc = fminf(fmaxf(x, 0.0f), (float)(Wd - 1));
            int   xl = (int)floorf(xc);
            int   xh = (xl + 1 < Wd - 1) ? (xl + 1) : (Wd - 1);
            float lx = xc - (float)xl;
            sXL[s] = xl; sXH[s] = xh;
            sWHX[s] = (1.0f - lx) * mf;
            sWLX[s] = lx * mf;
        } else if (t < NTAP) {
            // y sample (t-14): two tap rows j=2s, 2s+1; fold mask + 1/4 bin-avg
            int s = t - NS;
            float y1 = ry1 * scale - 0.5f, y2 = ry2 * scale - 0.5f;
            float g  = ((float)s + 0.5f) * (1.0f / PSR);
            float y  = y1 + g * (y2 - y1) * (1.0f / POUT);
            float mf = ((y > -1.0f) && (y < (float)Wd)) ? 1.0f : 0.0f;
            float yc = fminf(fmaxf(y, 0.0f), (float)(Wd - 1));
            int   yl = (int)floorf(yc);
            int   yh = (yl + 1 < Wd - 1) ? (yl + 1) : (Wd - 1);
            float ly = yc - (float)yl;
            sYROW[2 * s]     = yl;
            sYROW[2 * s + 1] = yh;
            sWY[2 * s]       = 0.25f * (1.0f - ly) * mf;
            sWY[2 * s + 1]   = 0.25f * ly * mf;
        } else {
            sMeta[0] = l;
            sMeta[1] = (int)roi[0];
            sMeta[2] = Wd;
        }
    }
    __syncthreads();

    const int l  = sMeta[0];
    const int b  = sMeta[1];
    const int Wd = sMeta[2];
    const float* fl = (l == 0) ? f0 : (l == 1) ? f1 : (l == 2) ? f2 : f3;
    const size_t plane = (size_t)Wd * (size_t)Wd;
    const float* fbase = fl + ((size_t)b * 256 + (size_t)ct * CT) * plane;

    // ---- Stage A: gather + x-interp tap rows into P[c][j][px] ----
    // 64 c * 28 j * 7 px = 12544 = 49 * 256 values, 4 gathers each.
    #pragma unroll 1
    for (int i = 0; i < 49; ++i) {
        int idx = i * 256 + t;                 // px fastest -> x-locality
        int c   = idx / 196;
        int r   = idx - c * 196;
        int j   = r / 7;
        int px  = r - j * 7;
        const float* rowp = fbase + (size_t)c * plane + (size_t)sYROW[j] * Wd;
        int s0 = 2 * px, s1 = s0 + 1;
        if (i == 0) __builtin_prefetch(rowp + sXL[s0], 0, 0);  // global_prefetch_b8
        float v = sWHX[s0] * rowp[sXL[s0]] + sWLX[s0] * rowp[sXH[s0]]
                + sWHX[s1] * rowp[sXL[s1]] + sWLX[s1] * rowp[sXH[s1]];
        sP[c * PSTR + j * 7 + px] = v;
    }
    __syncthreads();

    // ---- Stage B: y-reduction via chained V_WMMA_F32_16X16X4_F32 ----
    // Chain (sub16, px): D[16c x 16py] = sum_{py k-step} A[16c x 4tap] * B[4tap x 16py]
    // B is block diagonal: k-step py has nonzeros only in column py.
    const int wv   = t >> 5;
    const int lane = t & 31;
    const int n    = lane & 15;    // A row (c16) == B/D column (py)
    const int kh   = lane >> 4;    // k half: VGPR0 holds K = 2*kh, VGPR1 K = 2*kh+1
    for (int q = wv; q < (CT / 16) * POUT; q += 8) {
        int sub = q / POUT;
        int px  = q - sub * POUT;
        int cb  = sub * 16;
        v8f acc = {};
        #pragma unroll
        for (int py = 0; py < POUT; ++py) {
            int j0 = 4 * py + 2 * kh;
            int j1 = j0 + 1;
            v2f a, bm;
            a.x = sP[(cb + n) * PSTR + j0 * 7 + px];
            a.y = sP[(cb + n) * PSTR + j1 * 7 + px];
            float sel = (n == py) ? 1.0f : 0.0f;
            bm.x = sWY[j0] * sel;
            bm.y = sWY[j1] * sel;
            acc = __builtin_amdgcn_wmma_f32_16x16x4_f32(
                      false, a, false, bm, (short)0, acc, false, false);
        }
        // D: lane column N = n (py), VGPR r -> row M = r + 8*kh (channel in sub-tile)
        if (n < POUT) {
            float* op = out + ((size_t)m * 256 + (size_t)ct * CT + cb) * 49
                            + (size_t)n * 7 + px;
            #pragma unroll
            for (int r8 = 0; r8 < 8; ++r8) {
                op[(size_t)(r8 + 8 * kh) * 49] = acc[r8];
            }
        }
    }
}

extern "C" void kernel_launch(void* const* d_in, const int* in_sizes, int n_in,
                              void* d_out, int out_size, void* d_ws, size_t ws_size,
                              hipStream_t stream) {
    (void)n_in; (void)out_size; (void)d_ws; (void)ws_size;
    const float* f0   = (const float*)d_in[0];   // [2,256,256,256]
    const float* f1   = (const float*)d_in[1];   // [2,256,128,128]
    const float* f2   = (const float*)d_in[2];   // [2,256,64,64]
    const float* f3   = (const float*)d_in[3];   // [2,256,32,32]
    const float* rois = (const float*)d_in[4];   // [M,5]
    float* out = (float*)d_out;                  // [M,256,7,7]
    int M = in_sizes[4] / 5;
    dim3 grid((unsigned)M, 256 / CT);
    roi_align_fpn_wmma<<<grid, dim3(256), 0, stream>>>(f0, f1, f2, f3, rois, out, M);
}